// LIFmesoCell_55095840473308
// MI455X (gfx1250) — compile-verified
//
#include <hip/hip_runtime.h>
#include <hip/hip_bf16.h>

// ---- problem constants (from reference) ----
#define B_  256
#define M_  16
#define A_  4096
#define NS_ 1024
#define T_  64
#define AT_ (A_ + T_)            // Z_hist row length = 4160
#define DT_ 0.001f
#define DA_ 0.001f
#define REF_T_ 0.004f
#define N_NEUR_ 500.0f

// ---- flat output offsets (floats), tuple order:
// A_t, Z, Z_nll, lnPy, lnP1_y, V_t_new, snv_new, I_syn
#define BM_   ((size_t)B_ * M_)                       // 4096
#define BMNS_ ((size_t)B_ * M_ * NS_)                 // 4194304
#define BMA_  ((size_t)B_ * M_ * A_)                  // 16777216
#define OFF_AT   ((size_t)0)
#define OFF_Z    (OFF_AT  + BM_)
#define OFF_NLL  (OFF_Z   + BM_)
#define OFF_LNPY (OFF_NLL + BM_)
#define OFF_LNP1 (OFF_LNPY + BMNS_)
#define OFF_V    (OFF_LNP1 + BMNS_)
#define OFF_SNV  (OFF_V    + BMA_)
#define OFF_ISYN (OFF_SNV  + BMNS_)

typedef __attribute__((ext_vector_type(2))) float v2f;
typedef __attribute__((ext_vector_type(8))) float v8f;

// ---------------------------------------------------------------------------
// K1: I_syn update per (b,m).  clk branch resolved on device (graph-safe).
// ---------------------------------------------------------------------------
__global__ void lif_isyn_kernel(const float* __restrict__ I_syn,
                                const float* __restrict__ asyn,
                                const float* __restrict__ Zh,
                                const int*   __restrict__ clk_p,
                                float* __restrict__ ws_isyn,
                                float* __restrict__ out_isyn) {
    const int bm = blockIdx.x * blockDim.x + threadIdx.x;   // 0..4095
    if (bm >= (int)BM_) return;
    const int m = bm & (M_ - 1);
    const int clk = clk_p[0];
    float v = I_syn[bm];
    if (clk > 1) {  // SYN_DELAY_BINS == 1
        const float e = __expf(-DT_ * asyn[m]);
        const float zlast = Zh[(size_t)bm * AT_ + (clk + A_ - 2)];
        v = e * v + (1.0f - e) * zlast * (1.0f / DT_);
    }
    ws_isyn[bm] = v;
    out_isyn[bm] = v;
}

// ---------------------------------------------------------------------------
// K2: input_total = I_syn'(256x16) @ J(16x16) + x*eps, via V_WMMA_F32_16X16X4_F32.
// One wave per 16-row batch tile; K=16 as 4 chained WMMAs; C seeded with x*eps.
// A 16x4 layout (ISA 7.12.2): lanes 0-15 hold K={0,1}, lanes 16-31 hold K={2,3}.
// B 4x16: same K-half split across lane halves, row striped across lanes.
// C/D: VGPR j -> row j (lanes 0-15) / row j+8 (lanes 16-31), col = lane&15.
// ---------------------------------------------------------------------------
__global__ __launch_bounds__(32) void lif_gemm_wmma_kernel(
        const float* __restrict__ isyn_new,   // (B, M) row-major
        const float* __restrict__ J,          // (M, M) row-major, J[m][n]
        const float* __restrict__ x,          // (B, M)
        const float* __restrict__ eps_p,
        float* __restrict__ inp_tot) {        // (B, M)
    const int lane = threadIdx.x;
    const int half = lane >> 4;               // 0: lanes 0-15, 1: lanes 16-31
    const int lm   = lane & 15;               // N index (and M index for A-rows)
    const int rowbase = blockIdx.x * 16;      // batch tile
    const float eps = eps_p[0];

    v8f c;
#pragma unroll
    for (int j = 0; j < 8; ++j) {
        const int brow = rowbase + j + half * 8;
        c[j] = x[brow * M_ + lm] * eps;       // bias pre-loaded into accumulator
    }

#pragma unroll
    for (int kk = 0; kk < 4; ++kk) {
        const int k0 = kk * 4 + half * 2;
        v2f a, b;
        a[0] = isyn_new[(rowbase + lm) * M_ + k0];
        a[1] = isyn_new[(rowbase + lm) * M_ + k0 + 1];
        b[0] = J[(k0)     * M_ + lm];
        b[1] = J[(k0 + 1) * M_ + lm];
        // 8 args: (neg_a, A, neg_b, B, c_mod, C, reuse_a, reuse_b)
        c = __builtin_amdgcn_wmma_f32_16x16x4_f32(false, a, false, b,
                                                  (short)0, c, false, false);
    }

#pragma unroll
    for (int j = 0; j < 8; ++j) {
        const int brow = rowbase + j + half * 8;
        inp_tot[brow * M_ + lm] = c[j];
    }
}

// ---------------------------------------------------------------------------
// K3: main streaming pass over (b,m,a).  One block per (b,m), 256 threads,
// each thread 4x float4 = 16 elements.  Four simultaneous reductions.
// ---------------------------------------------------------------------------
__global__ __launch_bounds__(256) void lif_main_kernel(
        const float* __restrict__ V_old,
        const float* __restrict__ logS,
        const float* __restrict__ Zh,
        const float* __restrict__ amem,
        const float* __restrict__ rp,
        const float* __restrict__ ft,
        const float* __restrict__ inp_tot,
        const int*   __restrict__ clk_p,
        float* __restrict__ out) {
    const int bm = blockIdx.x;                 // 0..4095
    const int m  = bm & (M_ - 1);
    const int clk = clk_p[0];
    const float famem = amem[m];
    const float frp   = rp[m];
    const float fft   = ft[m];
    const float dinp  = DA_ * inp_tot[bm];

    const float* __restrict__ Vrow = V_old + (size_t)bm * A_;
    const float* __restrict__ Srow = logS  + (size_t)bm * A_;
    const float* __restrict__ Zrow = Zh    + (size_t)bm * AT_;
    float* __restrict__ Vout = out + OFF_V + (size_t)bm * A_;

    float zmac = 0.f, msum = 0.f, num = 0.f, den = 0.f;

#pragma unroll
    for (int it = 0; it < 4; ++it) {
        const int a = (threadIdx.x + it * 256) * 4;     // contiguous float4/lane
        const float4 v4 = *(const float4*)(Vrow + a);
        const float4 s4 = *(const float4*)(Srow + a);
        float vo[4] = {v4.x, v4.y, v4.z, v4.w};
        float sl[4] = {s4.x, s4.y, s4.z, s4.w};
        float vn[4];
#pragma unroll
        for (int j = 0; j < 4; ++j) {
            const int ai = a + j;
            const float age = (float)ai * DA_;
            float V = vo[j] + DA_ * (frp - vo[j]) * famem + dinp;
            if (age < REF_T_) V = 0.0f;
            const float lam  = (V == 0.0f) ? 1e-8f : __expf(V - fft);
            const float p    = lam * DT_;
            const float Plam = (p >= 0.01f) ? (1.0f - __expf(-p)) : p;
            const float zt   = Zrow[clk + (A_ - 1) - ai];       // reversed slice
            const float So   = __expf(sl[j]);
            const float mt   = So * zt;
            const float vt   = (1.0f - So) * mt;
            zmac += mt * Plam;
            msum += mt;
            num  += vt * Plam;
            den  += vt;
            vn[j] = V;
        }
        const float4 st = {vn[0], vn[1], vn[2], vn[3]};
        *(float4*)(Vout + a) = st;
    }

    // wave32 shuffle reduction, then 8-wave LDS combine
    for (int off = 16; off > 0; off >>= 1) {
        zmac += __shfl_down(zmac, off, 32);
        msum += __shfl_down(msum, off, 32);
        num  += __shfl_down(num,  off, 32);
        den  += __shfl_down(den,  off, 32);
    }
    __shared__ float sred[8][4];
    const int wave = threadIdx.x >> 5;
    if ((threadIdx.x & 31) == 0) {
        sred[wave][0] = zmac; sred[wave][1] = msum;
        sred[wave][2] = num;  sred[wave][3] = den;
    }
    __syncthreads();
    if (threadIdx.x == 0) {
        float Zm = 0.f, Ms = 0.f, Nu = 0.f, De = 0.f;
#pragma unroll
        for (int w = 0; w < 8; ++w) {
            Zm += sred[w][0]; Ms += sred[w][1];
            Nu += sred[w][2]; De += sred[w][3];
        }
        const float mess     = 1.0f - Ms;
        const float lambda_t = (De != 0.0f) ? (Nu / De) : 0.0f;
        float Z = Zm + lambda_t * mess;
        Z = fminf(fmaxf(Z, 0.0f), 1.0f);
        const float Ztn  = Zrow[clk + A_];
        const float var  = Z * (1.0f / N_NEUR_) + 0.1f * DT_ * (1.0f / N_NEUR_);
        const float diff = Ztn - Z;
        const float znll = 0.5f * diff * diff / var
                         + 0.5f * __logf(var)
                         + 0.5f * __logf(2.0f * 3.14f);
        out[OFF_AT  + bm] = Z * (1.0f / DT_);
        out[OFF_Z   + bm] = Z;
        out[OFF_NLL + bm] = znll;
    }
}

// ---------------------------------------------------------------------------
// K4: sampled-neuron elementwise pass over (b,m,ns).  One block per (b,m),
// 256 threads x float4 = 1024 = NS.
// ---------------------------------------------------------------------------
__global__ __launch_bounds__(256) void lif_sampled_kernel(
        const float* __restrict__ snv,
        const float* __restrict__ lastfire,
        const float* __restrict__ rp,
        const float* __restrict__ amem,
        const float* __restrict__ ft,
        const float* __restrict__ inp_tot,
        float* __restrict__ out) {
    const int bm = blockIdx.x;                 // 0..4095
    const int m  = bm & (M_ - 1);
    const float famem = amem[m];
    const float frp   = rp[m];
    const float fft   = ft[m];
    const float dinp  = DA_ * inp_tot[bm];

    const size_t base = (size_t)bm * NS_ + (size_t)threadIdx.x * 4;
    const float4 v4 = *(const float4*)(snv + base);
    const float4 g4 = *(const float4*)(lastfire + base);
    float vo[4] = {v4.x, v4.y, v4.z, v4.w};
    float ag[4] = {g4.x, g4.y, g4.z, g4.w};
    float o_v[4], o_py[4], o_p1[4];

#pragma unroll
    for (int j = 0; j < 4; ++j) {
        float V = vo[j] + DA_ * (frp - vo[j]) * famem + dinp;
        if (ag[j] < REF_T_) V = 0.0f;
        const float lam  = (V == 0.0f) ? 1e-8f : __expf(V - fft);
        const float ltdt = lam * DT_;
        const float Plam = (ltdt >= 0.01f) ? (1.0f - __expf(-ltdt)) : ltdt;
        o_v[j]  = V;
        o_py[j] = (ltdt < 0.01f) ? (V - fft + __logf(DT_)) : __logf(Plam);
        o_p1[j] = -ltdt;
    }
    *(float4*)(out + OFF_SNV  + base) = (float4){o_v[0],  o_v[1],  o_v[2],  o_v[3]};
    *(float4*)(out + OFF_LNPY + base) = (float4){o_py[0], o_py[1], o_py[2], o_py[3]};
    *(float4*)(out + OFF_LNP1 + base) = (float4){o_p1[0], o_p1[1], o_p1[2], o_p1[3]};
}

// ---------------------------------------------------------------------------
extern "C" void kernel_launch(void* const* d_in, const int* in_sizes, int n_in,
                              void* d_out, int out_size, void* d_ws, size_t ws_size,
                              hipStream_t stream) {
    const float* x        = (const float*)d_in[0];
    const float* V_t_old  = (const float*)d_in[1];
    const float* logS     = (const float*)d_in[2];
    const float* snv      = (const float*)d_in[3];
    const float* snlf     = (const float*)d_in[4];
    const float* I_syn    = (const float*)d_in[5];
    const float* amem     = (const float*)d_in[6];
    const float* asyn     = (const float*)d_in[7];
    const float* J        = (const float*)d_in[8];
    const float* rp       = (const float*)d_in[9];
    const float* ft       = (const float*)d_in[10];
    const float* eps      = (const float*)d_in[11];
    const float* Zh       = (const float*)d_in[12];
    const int*   clk      = (const int*)d_in[13];
    float* out = (float*)d_out;

    float* ws_isyn = (float*)d_ws;            // BM_ floats
    float* ws_inp  = ws_isyn + BM_;           // BM_ floats

    lif_isyn_kernel<<<(int)(BM_ / 256), 256, 0, stream>>>(
        I_syn, asyn, Zh, clk, ws_isyn, out + OFF_ISYN);

    lif_gemm_wmma_kernel<<<B_ / 16, 32, 0, stream>>>(
        ws_isyn, J, x, eps, ws_inp);

    lif_main_kernel<<<(int)BM_, 256, 0, stream>>>(
        V_t_old, logS, Zh, amem, rp, ft, ws_inp, clk, out);

    lif_sampled_kernel<<<(int)BM_, 256, 0, stream>>>(
        snv, snlf, rp, amem, ft, ws_inp, out);
}